// LinearCombiner_12876311953983
// MI455X (gfx1250) — compile-verified
//
#include <hip/hip_runtime.h>

typedef __attribute__((ext_vector_type(2))) float v2f;
typedef __attribute__((ext_vector_type(4))) float v4f;
typedef __attribute__((ext_vector_type(8))) float v8f;

constexpr int kD   = 1024;
constexpr int kHid = 3072;
constexpr int kOut = 8;
constexpr int kT   = 256;
constexpr int kU   = 128;

// Build X_text[t, 0:1024] = min(a,b), X_text[t, 1024:2048] = max(a,b)
__global__ __launch_bounds__(256) void prep_minmax_kernel(
    const float* __restrict__ a, const float* __restrict__ b,
    float* __restrict__ X)
{
    int i = blockIdx.x * 256 + threadIdx.x;
    if (i >= kT * kD) return;
    int t = i >> 10;          // / kD
    int d = i & (kD - 1);
    float av = a[i], bv = b[i];
    X[(size_t)t * (2 * kD) + d]       = fminf(av, bv);
    X[(size_t)t * (2 * kD) + kD + d]  = fmaxf(av, bv);
}

// C[M x N] = A[M x K] * (W[:, kofs:kofs+K])^T  (+ bias[N] if non-null)
// A row-major (stride K), W row-major (row stride ldw), C row-major (stride N).
// Wave tile = 16(M) x 64(N); 8 waves / block; K stepped by 4 via
// V_WMMA_F32_16X16X4_F32. Inner loop keeps TWO fragment groups in flight
// (depth-2 register pipeline) so WMMAs of group g wait only while groups
// g+1 and g+2 remain outstanding -> partial waits with ~10 loads in flight.
__global__ __launch_bounds__(256) void gemm_wmma_f32_kernel(
    const float* __restrict__ A, const float* __restrict__ W,
    const float* __restrict__ bias, float* __restrict__ C,
    int M, int N, int K, int ldw, int kofs)
{
    const int lane = threadIdx.x & 31;
    const int wid  = threadIdx.x >> 5;
    const int half = lane >> 4;       // which 16-lane half of the wave
    const int lrow = lane & 15;
    const int nsup = N >> 6;          // 64-wide supertiles in N
    const int total = (M >> 4) * nsup;
    const int wt = blockIdx.x * 8 + wid;
    if (wt >= total) return;          // uniform per wave -> EXEC stays all-ones
    const int m0 = (wt / nsup) << 4;
    const int n0 = (wt % nsup) << 6;

    // A fragment (16x4 f32): lane (half,lrow) holds A[m0+lrow, k+2*half .. +1]
    const float* Ap  = A + (size_t)(m0 + lrow) * K + 2 * half;
    // B fragment (4x16 f32 = W^T tile): lane holds W[n0+lrow, kofs+k+2*half .. +1]
    const float* Wp0 = W + (size_t)(n0 + lrow) * ldw + kofs + 2 * half;
    const float* Wp1 = Wp0 + (size_t)16 * ldw;
    const float* Wp2 = Wp0 + (size_t)32 * ldw;
    const float* Wp3 = Wp0 + (size_t)48 * ldw;

    v8f acc0 = {}, acc1 = {}, acc2 = {}, acc3 = {};

    // Pipeline prologue: two groups in flight (k = 0 and k = 4)
    v2f aA  = *(const v2f*)(Ap);
    v2f aB0 = *(const v2f*)(Wp0);
    v2f aB1 = *(const v2f*)(Wp1);
    v2f aB2 = *(const v2f*)(Wp2);
    v2f aB3 = *(const v2f*)(Wp3);

    v2f nA  = *(const v2f*)(Ap  + 4);
    v2f nB0 = *(const v2f*)(Wp0 + 4);
    v2f nB1 = *(const v2f*)(Wp1 + 4);
    v2f nB2 = *(const v2f*)(Wp2 + 4);
    v2f nB3 = *(const v2f*)(Wp3 + 4);

#pragma unroll 4
    for (int k = 8; k < K; k += 4) {
        // Issue group k (consumed two iterations from now)
        v2f fA  = *(const v2f*)(Ap  + k);
        v2f fB0 = *(const v2f*)(Wp0 + k);
        v2f fB1 = *(const v2f*)(Wp1 + k);
        v2f fB2 = *(const v2f*)(Wp2 + k);
        v2f fB3 = *(const v2f*)(Wp3 + k);
        // Matrix work on the OLDEST group (two newer groups stay outstanding)
        acc0 = __builtin_amdgcn_wmma_f32_16x16x4_f32(false, aA, false, aB0, (short)0, acc0, false, false);
        acc1 = __builtin_amdgcn_wmma_f32_16x16x4_f32(false, aA, false, aB1, (short)0, acc1, false, false);
        acc2 = __builtin_amdgcn_wmma_f32_16x16x4_f32(false, aA, false, aB2, (short)0, acc2, false, false);
        acc3 = __builtin_amdgcn_wmma_f32_16x16x4_f32(false, aA, false, aB3, (short)0, acc3, false, false);
        // Rotate the pipeline
        aA = nA; aB0 = nB0; aB1 = nB1; aB2 = nB2; aB3 = nB3;
        nA = fA; nB0 = fB0; nB1 = fB1; nB2 = fB2; nB3 = fB3;
    }
    // Pipeline epilogue: drain the two remaining groups
    acc0 = __builtin_amdgcn_wmma_f32_16x16x4_f32(false, aA, false, aB0, (short)0, acc0, false, false);
    acc1 = __builtin_amdgcn_wmma_f32_16x16x4_f32(false, aA, false, aB1, (short)0, acc1, false, false);
    acc2 = __builtin_amdgcn_wmma_f32_16x16x4_f32(false, aA, false, aB2, (short)0, acc2, false, false);
    acc3 = __builtin_amdgcn_wmma_f32_16x16x4_f32(false, aA, false, aB3, (short)0, acc3, false, false);
    acc0 = __builtin_amdgcn_wmma_f32_16x16x4_f32(false, nA, false, nB0, (short)0, acc0, false, false);
    acc1 = __builtin_amdgcn_wmma_f32_16x16x4_f32(false, nA, false, nB1, (short)0, acc1, false, false);
    acc2 = __builtin_amdgcn_wmma_f32_16x16x4_f32(false, nA, false, nB2, (short)0, acc2, false, false);
    acc3 = __builtin_amdgcn_wmma_f32_16x16x4_f32(false, nA, false, nB3, (short)0, acc3, false, false);

    const float bc0 = bias ? bias[n0      + lrow] : 0.0f;
    const float bc1 = bias ? bias[n0 + 16 + lrow] : 0.0f;
    const float bc2 = bias ? bias[n0 + 32 + lrow] : 0.0f;
    const float bc3 = bias ? bias[n0 + 48 + lrow] : 0.0f;

    // C/D layout: VGPR v -> row m0 + v + 8*half, column n0 + lrow
#pragma unroll
    for (int v = 0; v < 8; ++v) {
        float* cr = C + (size_t)(m0 + v + 8 * half) * N + n0 + lrow;
        cr[0]  = acc0[v] + bc0;
        cr[16] = acc1[v] + bc1;
        cr[32] = acc2[v] + bc2;
        cr[48] = acc3[v] + bc3;
    }
}

// Narrow final layer: C[M x N] = A[M x K] * W^T + bias, N small (8).
// One wave per output element: vectorized strided partials + wave reduction.
__global__ __launch_bounds__(256) void gemm_smallN_kernel(
    const float* __restrict__ A, const float* __restrict__ W,
    const float* __restrict__ bias, float* __restrict__ C,
    int M, int K, int N)
{
    const int lane = threadIdx.x & 31;
    const int wid  = threadIdx.x >> 5;
    const int idx  = blockIdx.x * 8 + wid;
    if (idx >= M * N) return;
    const int r = idx / N;
    const int o = idx - r * N;
    const v4f* a4 = (const v4f*)(A + (size_t)r * K);
    const v4f* w4 = (const v4f*)(W + (size_t)o * K);
    const int n4 = K >> 2;
    float s = 0.0f;
    for (int k = lane; k < n4; k += 32) {
        v4f x = a4[k], y = w4[k];
        s += x.x * y.x + x.y * y.y + x.z * y.z + x.w * y.w;
    }
#pragma unroll
    for (int off = 16; off > 0; off >>= 1)
        s += __shfl_down(s, off, 32);
    if (lane == 0) C[idx] = s + (bias ? bias[o] : 0.0f);
}

// out[t,u,o] = Ctext[t,o] + Cuser[u,o]
__global__ __launch_bounds__(256) void combine_kernel(
    const float* __restrict__ Ct, const float* __restrict__ Cu,
    float* __restrict__ out)
{
    int i = blockIdx.x * 256 + threadIdx.x;
    if (i >= kT * kU * kOut) return;
    int o = i & (kOut - 1);
    int u = (i >> 3) & (kU - 1);
    int t = i >> 10;
    out[i] = Ct[t * kOut + o] + Cu[u * kOut + o];
}

extern "C" void kernel_launch(void* const* d_in, const int* in_sizes, int n_in,
                              void* d_out, int out_size, void* d_ws, size_t ws_size,
                              hipStream_t stream) {
    (void)in_sizes; (void)n_in; (void)out_size; (void)ws_size;
    const float* text_a = (const float*)d_in[0];
    const float* text_b = (const float*)d_in[1];
    const float* user   = (const float*)d_in[2];
    const float* W1     = (const float*)d_in[3];
    const float* b1     = (const float*)d_in[4];
    const float* W2     = (const float*)d_in[5];
    const float* b2     = (const float*)d_in[6];
    const float* W3     = (const float*)d_in[7];
    const float* b3     = (const float*)d_in[8];
    float* out = (float*)d_out;

    // Workspace carve-up (~11.0 MB total, fp32)
    float* ws    = (float*)d_ws;
    float* Xtext = ws; ws += (size_t)kT * 2 * kD;   // 256 x 2048
    float* Ytext = ws; ws += (size_t)kT * kHid;     // 256 x 3072
    float* Yuser = ws; ws += (size_t)kU * kHid;     // 128 x 3072
    float* Ztext = ws; ws += (size_t)kT * kHid;     // 256 x 3072
    float* Zuser = ws; ws += (size_t)kU * kHid;     // 128 x 3072
    float* Ctext = ws; ws += (size_t)kT * kOut;     // 256 x 8
    float* Cuser = ws; ws += (size_t)kU * kOut;     // 128 x 8

    // Stage 0: min/max concat for the text branch
    prep_minmax_kernel<<<(kT * kD + 255) / 256, 256, 0, stream>>>(text_a, text_b, Xtext);

    auto gemm_blocks = [](int M, int N) { return ((M >> 4) * (N >> 6) + 7) / 8; };

    // Layer 1 (split over concat): Ytext = Xtext @ W1[:, :2048]^T + b1 ; Yuser = user @ W1[:, 2048:]^T
    gemm_wmma_f32_kernel<<<gemm_blocks(kT, kHid), 256, 0, stream>>>(
        Xtext, W1, b1, Ytext, kT, kHid, 2 * kD, kHid, 0);
    gemm_wmma_f32_kernel<<<gemm_blocks(kU, kHid), 256, 0, stream>>>(
        user, W1, nullptr, Yuser, kU, kHid, kD, kHid, 2 * kD);

    // Layer 2 (distributes over the sum): Ztext = Ytext @ W2^T + b2 ; Zuser = Yuser @ W2^T
    gemm_wmma_f32_kernel<<<gemm_blocks(kT, kHid), 256, 0, stream>>>(
        Ytext, W2, b2, Ztext, kT, kHid, kHid, kHid, 0);
    gemm_wmma_f32_kernel<<<gemm_blocks(kU, kHid), 256, 0, stream>>>(
        Yuser, W2, nullptr, Zuser, kU, kHid, kHid, kHid, 0);

    // Layer 3 (N=8): Ctext = Ztext @ W3^T + b3 ; Cuser = Zuser @ W3^T
    gemm_smallN_kernel<<<(kT * kOut + 7) / 8, 256, 0, stream>>>(
        Ztext, W3, b3, Ctext, kT, kHid, kOut);
    gemm_smallN_kernel<<<(kU * kOut + 7) / 8, 256, 0, stream>>>(
        Zuser, W3, nullptr, Cuser, kU, kHid, kOut);

    // Broadcast combine: out[t,u,:] = Ctext[t,:] + Cuser[u,:]
    combine_kernel<<<(kT * kU * kOut + 255) / 256, 256, 0, stream>>>(Ctext, Cuser, out);
}